// PointNet2Encoder_8950711845080
// MI455X (gfx1250) — compile-verified
//
#include <hip/hip_runtime.h>
#include <math.h>

#define BN_EPS 1e-5f

typedef __attribute__((ext_vector_type(2))) float v2f;
typedef __attribute__((ext_vector_type(8))) float v8f;

// ---------------------------------------------------------------------------
// Transpose xyz (B,9,N) -> pts (B,N,9) and xyz3 (B,N,3)
// ---------------------------------------------------------------------------
__global__ void k_transpose(const float* __restrict__ xin, float* __restrict__ pts,
                            float* __restrict__ xyz3, int B, int N) {
  int t = blockIdx.x * blockDim.x + threadIdx.x;
  if (t >= B * N) return;
  int b = t / N, n = t % N;
  const float* src = xin + (size_t)b * 9 * N + n;
  float* pd = pts + (size_t)t * 9;
#pragma unroll
  for (int c = 0; c < 9; ++c) pd[c] = src[(size_t)c * N];
  float* xd = xyz3 + (size_t)t * 3;
  xd[0] = pd[0]; xd[1] = pd[1]; xd[2] = pd[2];
}

// ---------------------------------------------------------------------------
// Per-conv prep: pad W to (ColPad x Kd) zero-filled, fold BN into
// scale/shift arrays (zero-padded to ColPad).
// ---------------------------------------------------------------------------
__global__ void k_prep(const float* __restrict__ W, const float* __restrict__ bias,
                       const float* __restrict__ gamma, const float* __restrict__ beta,
                       const float* __restrict__ mu, const float* __restrict__ var,
                       int Cout, int KW, int Kd, int ColPad,
                       float* __restrict__ Wpad, float* __restrict__ scale,
                       float* __restrict__ shift) {
  int t = blockIdx.x * blockDim.x + threadIdx.x;
  int totalW = ColPad * Kd;
  if (t < totalW) {
    int col = t / Kd, c = t % Kd;
    Wpad[t] = (col < Cout && c < KW) ? W[col * KW + c] : 0.0f;
  }
  int u = t - totalW;
  if (u >= 0 && u < ColPad) {
    if (u < Cout) {
      float s0 = gamma[u] * rsqrtf(var[u] + BN_EPS);
      scale[u] = s0;
      shift[u] = bias[u] * s0 + beta[u] - mu[u] * s0;
    } else {
      scale[u] = 0.0f;
      shift[u] = 0.0f;
    }
  }
}

// ---------------------------------------------------------------------------
// Farthest point sampling: one block per batch. Distance array in LDS,
// argmax via LDS tree reduction (first-index tie break, like jnp.argmax).
// ---------------------------------------------------------------------------
__global__ void k_fps(const float* __restrict__ xyz, int N, int np,
                      int* __restrict__ fidx) {
  extern __shared__ float smem[];
  float* sd = smem;                       // N
  float* rv = sd + N;                     // blockDim
  int* ri = (int*)(rv + blockDim.x);      // blockDim
  int b = blockIdx.x, tid = threadIdx.x, nt = blockDim.x;
  const float* xb = xyz + (size_t)b * N * 3;
  for (int n = tid; n < N; n += nt) sd[n] = 1e10f;
  __syncthreads();
  int far = 0;
  int* fb = fidx + (size_t)b * np;
  for (int it = 0; it < np; ++it) {
    if (tid == 0) fb[it] = far;
    float fx = xb[far * 3 + 0], fy = xb[far * 3 + 1], fz = xb[far * 3 + 2];
    float bestV = -1.0f; int bestI = 0;
    for (int n = tid; n < N; n += nt) {
      float dx = xb[n * 3 + 0] - fx;
      float dy = xb[n * 3 + 1] - fy;
      float dz = xb[n * 3 + 2] - fz;
      float d = dx * dx + dy * dy + dz * dz;
      float dm = fminf(sd[n], d);
      sd[n] = dm;
      if (dm > bestV) { bestV = dm; bestI = n; }
    }
    rv[tid] = bestV; ri[tid] = bestI;
    __syncthreads();
    for (int s = nt >> 1; s > 0; s >>= 1) {
      if (tid < s) {
        float ov = rv[tid + s]; int oi = ri[tid + s];
        if (ov > rv[tid] || (ov == rv[tid] && oi < ri[tid])) {
          rv[tid] = ov; ri[tid] = oi;
        }
      }
      __syncthreads();
    }
    far = ri[0];
    __syncthreads();
  }
}

// ---------------------------------------------------------------------------
// Gather new_xyz = xyz[fps_idx]
// ---------------------------------------------------------------------------
__global__ void k_gather_newxyz(const float* __restrict__ xyz,
                                const int* __restrict__ fidx,
                                float* __restrict__ nxyz, int N, int np, int total) {
  int t = blockIdx.x * blockDim.x + threadIdx.x;
  if (t >= total) return;
  int b = t / np;
  int i = fidx[t];
  const float* s = xyz + ((size_t)b * N + i) * 3;
  float* d = nxyz + (size_t)t * 3;
  d[0] = s[0]; d[1] = s[1]; d[2] = s[2];
}

// ---------------------------------------------------------------------------
// Ball query: first Kg indices with d2 <= r2, padded with first hit.
// ---------------------------------------------------------------------------
__global__ void k_ballq(const float* __restrict__ xyz, const float* __restrict__ nxyz,
                        int N, int np, int Kg, float r2, int* __restrict__ gidx,
                        int total) {
  int t = blockIdx.x * blockDim.x + threadIdx.x;
  if (t >= total) return;
  int b = t / np;
  const float* c = nxyz + (size_t)t * 3;
  float cx = c[0], cy = c[1], cz = c[2];
  const float* xb = xyz + (size_t)b * N * 3;
  int* g = gidx + (size_t)t * Kg;
  int cnt = 0, first = 0; bool found = false;
  for (int n = 0; n < N && cnt < Kg; ++n) {
    float dx = xb[n * 3 + 0] - cx;
    float dy = xb[n * 3 + 1] - cy;
    float dz = xb[n * 3 + 2] - cz;
    float d2 = dx * dx + dy * dy + dz * dz;
    if (d2 <= r2) {
      if (!found) { found = true; first = n; }
      g[cnt++] = n;
    }
  }
  for (; cnt < Kg; ++cnt) g[cnt] = first;
}

// ---------------------------------------------------------------------------
// Fused branch kernel: gather grouped features into LDS, run the 3 pointwise
// convs (V_WMMA_F32_16X16X4_F32 + padded-BN epilogue) through LDS ping-pong,
// then max-over-K. One workgroup per (b, s) group. Inner loop is branch-free:
// ds_load_b64 (A) + global_load_b64 (Wpad B) + wmma.
// ---------------------------------------------------------------------------
struct ConvP {
  const float* Wpad;    // (ColPad x Kdim), zero-padded
  const float* scale;   // ColPad, zero-padded
  const float* shift;   // ColPad, zero-padded
  int Cout;
  int Kdim;             // padded K extent (multiple of 4)
};

struct BranchA {
  const float* pts; int Cpts;   // feature source (B, Npts, Cpts)
  const float* xyz;             // (B, Npts, 3)
  const float* nxyz;            // (B, np, 3)
  const int* gidx;              // (B, np, Kg)
  float* out; int outStride, outOff;
  int Npts, np, Kg;
  int Cin;                      // Cpts + 3
  int Cin_pad;                  // Cin rounded up to multiple of 4
  int stridePad;                // LDS row stride (floats, multiple of 4)
  ConvP conv[3];
};

__global__ void k_branch(BranchA a) {
  extern __shared__ float lds[];
  float* bufA = lds;
  float* bufB = lds + (size_t)a.Kg * a.stridePad;
  int g = blockIdx.x;                 // (b, s)
  int b = g / a.np;
  int tid = threadIdx.x;
  int lane = tid & 31, wave = tid >> 5, nwaves = blockDim.x >> 5;

  const int* gi = a.gidx + (size_t)g * a.Kg;
  const float* cen = a.nxyz + (size_t)g * 3;
  float cx = cen[0], cy = cen[1], cz = cen[2];

  // 1. gather grouped features: [points feats | rel-xyz | zero pad]
  int totalElems = a.Kg * a.Cin_pad;
  for (int e = tid; e < totalElems; e += blockDim.x) {
    int k = e / a.Cin_pad, c = e % a.Cin_pad;
    int idx = gi[k];
    float v = 0.0f;
    if (c < a.Cpts) {
      v = a.pts[((size_t)b * a.Npts + idx) * a.Cpts + c];
    } else if (c < a.Cin) {
      const float* p = a.xyz + ((size_t)b * a.Npts + idx) * 3;
      int cc = c - a.Cpts;
      v = p[cc] - (cc == 0 ? cx : (cc == 1 ? cy : cz));
    }
    bufA[k * a.stridePad + c] = v;
  }
  __syncthreads();

  float* cur = bufA;
  float* nxt = bufB;
  int nRowT = a.Kg >> 4;
  int lsub = lane & 15;
  int khalf = (lane < 16) ? 0 : 2;   // A/B fragment K offset per lane half
  int rowoff = (lane < 16) ? 0 : 8;  // C/D fragment row offset per lane half

  for (int ci = 0; ci < 3; ++ci) {
    ConvP cp = a.conv[ci];
    int nColT = (cp.Cout + 15) >> 4;
    int nT = nRowT * nColT;
    int Kd = cp.Kdim;
    for (int t = wave; t < nT; t += nwaves) {
      int rt = t / nColT, ct = t % nColT;
      v8f acc = {0.f, 0.f, 0.f, 0.f, 0.f, 0.f, 0.f, 0.f};
      const float* Arow = cur + (rt * 16 + lsub) * a.stridePad + khalf;
      int col = ct * 16 + lsub;
      const float* Wrow = cp.Wpad + (size_t)col * Kd + khalf;
#pragma unroll 4
      for (int kk = 0; kk < Kd; kk += 4) {
        v2f av = *(const v2f*)(Arow + kk);   // ds_load_b64
        v2f bv = *(const v2f*)(Wrow + kk);   // global_load_b64, unconditional
        acc = __builtin_amdgcn_wmma_f32_16x16x4_f32(
            false, av, false, bv, (short)0, acc, false, false);
      }
      float sc = cp.scale[col];
      float sh = cp.shift[col];
#pragma unroll
      for (int v = 0; v < 8; ++v) {
        int row = rt * 16 + v + rowoff;
        nxt[row * a.stridePad + col] = fmaxf(acc[v] * sc + sh, 0.0f);
      }
    }
    __syncthreads();
    float* tmp = cur; cur = nxt; nxt = tmp;
  }

  // 2. max over the Kg grouped points, write to concatenated output channels
  int Cl = a.conv[2].Cout;
  for (int o = tid; o < Cl; o += blockDim.x) {
    float m = cur[o];
    for (int k = 1; k < a.Kg; ++k) m = fmaxf(m, cur[k * a.stridePad + o]);
    a.out[(size_t)g * a.outStride + a.outOff + o] = m;
  }
}

// ---------------------------------------------------------------------------
// VAE head: x = (mean + exp(0.5*clip(logv))*noise)*0.18215 ; plus xyz output
// ---------------------------------------------------------------------------
__global__ void k_head(const float* __restrict__ lpts, const float* __restrict__ nxyz,
                       const float* __restrict__ noise, float* __restrict__ out) {
  int t = blockIdx.x * blockDim.x + threadIdx.x;  // b*64 + n
  if (t >= 16 * 64) return;
  int b = t / 64, n = t % 64;
  const float* f = lpts + (size_t)t * 8;
#pragma unroll
  for (int c = 0; c < 4; ++c) {
    float mean = f[c];
    float logv = f[4 + c];
    logv = fminf(fmaxf(logv, -30.f), 20.f);
    float nz = noise[((size_t)b * 4 + c) * 64 + n];
    out[((size_t)b * 4 + c) * 64 + n] = (mean + expf(0.5f * logv) * nz) * 0.18215f;
  }
  const float* x = nxyz + (size_t)t * 3;
#pragma unroll
  for (int c = 0; c < 3; ++c)
    out[16 * 4 * 64 + ((size_t)b * 3 + c) * 64 + n] = x[c];
}

// ---------------------------------------------------------------------------
extern "C" void kernel_launch(void* const* d_in, const int* in_sizes, int n_in,
                              void* d_out, int out_size, void* d_ws, size_t ws_size,
                              hipStream_t stream) {
  (void)in_sizes; (void)n_in; (void)out_size; (void)ws_size;
  const float* xin = (const float*)d_in[0];
  const float* noise = (const float*)d_in[1];
  auto P = [&](int layer, int branch, int conv, int t) -> const float* {
    return (const float*)d_in[2 + (((layer * 2) + branch) * 3 + conv) * 6 + t];
  };
  const int B = 16, N0 = 8192;

  char* ws = (char*)d_ws;
  size_t off = 0;
  auto alloc = [&](size_t bytes) -> void* {
    void* p = ws + off;
    off = (off + bytes + 255) & ~(size_t)255;
    return p;
  };
  float* pts0 = (float*)alloc((size_t)B * N0 * 9 * 4);
  float* xyz0 = (float*)alloc((size_t)B * N0 * 3 * 4);
  int* fidx = (int*)alloc((size_t)B * 1024 * 4);
  int* gidx = (int*)alloc((size_t)B * 1024 * 32 * 4);
  const int NP[4] = {1024, 512, 256, 64};
  const int CT[4] = {96, 256, 512, 8};
  float* nxyz[4]; float* lpts[4];
  for (int l = 0; l < 4; ++l) {
    nxyz[l] = (float*)alloc((size_t)B * NP[l] * 3 * 4);
    lpts[l] = (float*)alloc((size_t)B * NP[l] * CT[l] * 4);
  }

  const float RAD[4][2] = {{0.05f, 0.1f}, {0.1f, 0.2f}, {0.2f, 0.4f}, {0.4f, 0.8f}};
  const int KG[2] = {16, 32};
  const int MLP[4][2][3] = {{{16, 16, 32}, {32, 32, 64}},
                            {{64, 64, 128}, {64, 96, 128}},
                            {{128, 196, 256}, {128, 196, 256}},
                            {{256, 256, 4}, {256, 384, 4}}};
  const int CIN[4] = {12, 99, 259, 515};      // curC + 3
  const int CINPAD[4] = {12, 100, 260, 516};  // align4

  // --- weight prep: pad W, fold BN, per conv ---
  ConvP convs[4][2][3];
  for (int l = 0; l < 4; ++l) {
    for (int br = 0; br < 2; ++br) {
      for (int ci = 0; ci < 3; ++ci) {
        int Cout = MLP[l][br][ci];
        int KW = (ci == 0) ? CIN[l] : MLP[l][br][ci - 1];
        int Kd = (ci == 0) ? CINPAD[l] : MLP[l][br][ci - 1];
        int ColPad = (Cout + 15) & ~15;
        float* Wpad = (float*)alloc((size_t)ColPad * Kd * 4);
        float* scale = (float*)alloc((size_t)ColPad * 4);
        float* shift = (float*)alloc((size_t)ColPad * 4);
        int total = ColPad * Kd + ColPad;
        k_prep<<<(total + 255) / 256, 256, 0, stream>>>(
            P(l, br, ci, 0), P(l, br, ci, 1), P(l, br, ci, 2), P(l, br, ci, 3),
            P(l, br, ci, 4), P(l, br, ci, 5), Cout, KW, Kd, ColPad, Wpad, scale,
            shift);
        convs[l][br][ci].Wpad = Wpad;
        convs[l][br][ci].scale = scale;
        convs[l][br][ci].shift = shift;
        convs[l][br][ci].Cout = Cout;
        convs[l][br][ci].Kdim = Kd;
      }
    }
  }

  {
    int total = B * N0;
    k_transpose<<<(total + 255) / 256, 256, 0, stream>>>(xin, pts0, xyz0, B, N0);
  }

  const float* curXyz = xyz0;
  const float* curPts = pts0;
  int curN = N0, curC = 9;

  for (int l = 0; l < 4; ++l) {
    int np = NP[l];
    {
      size_t lds = (size_t)curN * 4 + (size_t)1024 * 8;
      k_fps<<<B, 1024, lds, stream>>>(curXyz, curN, np, fidx);
    }
    {
      int total = B * np;
      k_gather_newxyz<<<(total + 255) / 256, 256, 0, stream>>>(curXyz, fidx, nxyz[l],
                                                               curN, np, total);
    }
    int chOff = 0;
    for (int br = 0; br < 2; ++br) {
      int Kg = KG[br];
      float r = RAD[l][br];
      int total = B * np;
      k_ballq<<<(total + 255) / 256, 256, 0, stream>>>(curXyz, nxyz[l], curN, np, Kg,
                                                       r * r, gidx, total);
      BranchA a;
      a.pts = curPts; a.Cpts = curC;
      a.xyz = curXyz; a.nxyz = nxyz[l]; a.gidx = gidx;
      a.out = lpts[l]; a.outStride = CT[l]; a.outOff = chOff;
      a.Npts = curN; a.np = np; a.Kg = Kg;
      a.Cin = CIN[l];
      a.Cin_pad = CINPAD[l];
      int maxC = a.Cin_pad;
      for (int ci = 0; ci < 3; ++ci) {
        a.conv[ci] = convs[l][br][ci];
        if (a.conv[ci].Cout > maxC) maxC = a.conv[ci].Cout;
      }
      a.stridePad = maxC + 4;  // multiple of 4; >= every ColPad; staggers banks
      size_t lds = (size_t)2 * Kg * a.stridePad * 4;
      k_branch<<<B * np, 256, lds, stream>>>(a);
      chOff += MLP[l][br][2];
    }
    curXyz = nxyz[l]; curPts = lpts[l];
    curN = np; curC = CT[l];
  }

  k_head<<<(B * 64 + 255) / 256, 256, 0, stream>>>(lpts[3], nxyz[3], noise,
                                                   (float*)d_out);
}